// SlidingWindowDropout2D_48790828482806
// MI455X (gfx1250) — compile-verified
//
#include <hip/hip_runtime.h>
#include <cstdint>

#define B_   16
#define C_   64
#define H_   256
#define W_   256
#define HC_  250
#define WC_  250
#define WSZ  7
#define DROPP 0.1f
#define PLANES (B_ * C_)        // 1024
#define HW     (H_ * W_)        // 65536
#define ROWS_PER_BLK 4
#define PPB    32               // planes processed per block
#define GRID_Y (PLANES / PPB)   // 32

typedef float v4f __attribute__((ext_vector_type(4)));

// mask(h,w) = 0 if any anchor (ah,aw) in [h-6,h]x[w-6,w] (clipped to [0,249])
// has u[ah,aw] < 0.1, else 1.  (== 1 - maxpool7x7(pad6(decisions)))
__device__ __forceinline__ float mask_at(const float* __restrict__ u, int h, int w) {
    int ah0 = h - (WSZ - 1); if (ah0 < 0) ah0 = 0;
    int ah1 = h;             if (ah1 > HC_ - 1) ah1 = HC_ - 1;
    int aw0 = w - (WSZ - 1); if (aw0 < 0) aw0 = 0;
    int aw1 = w;             if (aw1 > WC_ - 1) aw1 = WC_ - 1;
    int dropped = 0;
    for (int ah = ah0; ah <= ah1; ++ah) {
        const float* row = u + ah * WC_;
        for (int aw = aw0; aw <= aw1; ++aw)
            dropped |= (row[aw] < DROPP) ? 1 : 0;
    }
    return dropped ? 0.0f : 1.0f;
}

// Kernel 1: materialize the [256,256] mask (tiny; stays L2-resident afterwards).
__global__ __launch_bounds__(256) void swdrop_build_mask(const float* __restrict__ u,
                                                         float* __restrict__ mask) {
    int idx = blockIdx.x * 256 + threadIdx.x;   // 0..65535
    int h = idx >> 8;
    int w = idx & 255;
    mask[idx] = mask_at(u, h, w);
}

// Kernel 2: stream x -> out. Block = 256 threads = 4 mask rows x 64 float4 cols,
// iterating over PPB planes with the mask held in registers. x/out use NT hints
// (no reuse; don't churn the 192 MB L2), mask stays regular-temporal.
__global__ __launch_bounds__(256) void swdrop_apply(const float* __restrict__ x,
                                                    float* __restrict__ out,
                                                    const float* __restrict__ mask,
                                                    const float* __restrict__ u) {
    __shared__ float sm[ROWS_PER_BLK * W_];     // 4 KB mask tile
    const int t  = threadIdx.x;
    const int c  = (t & 63) << 2;               // column (float index), 0..252 step 4
    const int rl = t >> 6;                      // local row 0..3
    const int h  = blockIdx.x * ROWS_PER_BLK + rl;

    if (mask != nullptr) {
        // CDNA5 async global->LDS staging of the 4 KB mask tile (16 B / lane).
        const float* gp = mask + (size_t)blockIdx.x * (ROWS_PER_BLK * W_) + (size_t)t * 4;
        unsigned ldsa = (unsigned)(uintptr_t)(&sm[t * 4]);   // flat low 32 bits == LDS offset
        asm volatile("global_load_async_to_lds_b128 %0, %1, off"
                     :: "v"(ldsa), "v"(gp) : "memory");
        asm volatile("s_wait_asynccnt 0x0" ::: "memory");
    } else {
        // Fallback: compute this block's mask tile in-place (ws too small).
        v4f m;
        m.x = mask_at(u, h, c + 0);
        m.y = mask_at(u, h, c + 1);
        m.z = mask_at(u, h, c + 2);
        m.w = mask_at(u, h, c + 3);
        *reinterpret_cast<v4f*>(&sm[t * 4]) = m;
    }
    __syncthreads();

    const v4f mv = reinterpret_cast<const v4f*>(sm)[t];

    const size_t stride4 = HW >> 2;             // float4 per plane (16384)
    size_t idx4 = (size_t)(blockIdx.y * PPB) * stride4 + (size_t)((h * W_ + c) >> 2);
    const v4f* xi = reinterpret_cast<const v4f*>(x) + idx4;
    v4f*       oo = reinterpret_cast<v4f*>(out) + idx4;

#pragma unroll 8
    for (int p = 0; p < PPB; ++p) {
        v4f v = __builtin_nontemporal_load(xi + (size_t)p * stride4);
        v *= mv;
        __builtin_nontemporal_store(v, oo + (size_t)p * stride4);
    }
}

extern "C" void kernel_launch(void* const* d_in, const int* in_sizes, int n_in,
                              void* d_out, int out_size, void* d_ws, size_t ws_size,
                              hipStream_t stream) {
    (void)in_sizes; (void)n_in; (void)out_size;
    const float* x = (const float*)d_in[0];
    const float* u = (const float*)d_in[1];
    float* out = (float*)d_out;

    float* mask = nullptr;
    if (d_ws != nullptr && ws_size >= (size_t)HW * sizeof(float)) {
        mask = (float*)d_ws;
        swdrop_build_mask<<<HW / 256, 256, 0, stream>>>(u, mask);
    }
    swdrop_apply<<<dim3(H_ / ROWS_PER_BLK, GRID_Y), 256, 0, stream>>>(x, out, mask, u);
}